// FrFTLayer_17806934409597
// MI455X (gfx1250) — compile-verified
//
#include <hip/hip_runtime.h>
#include <stdint.h>

// ---------------------------------------------------------------------------
// FrFT (Ozaktas, 0.5<a<1.5) for x[256][4096] -> complex64[256][4096].
// Both convolution stages are Toeplitz, evaluated as WMMA f32 GEMMs.
// Stage-2 chirp table window is staged into LDS via async global->LDS loads.
// ---------------------------------------------------------------------------

#define NN   4096          // transform length
#define RR   256           // rows
#define JP   8192          // padded width of g (2N-1 = 8191 -> 8192, last col zero)
#define WLEN 16384         // w-table complex entries (valid 0..16380)
#define SLEN 8192          // sinc-table entries (valid 0..8190)

#define TJ   256           // K-chunk per LDS stage (stage-2)
#define SPAN 768           // LDS window entries >= 510 + TJ = 766

// workspace layout (float offsets)
#define WTAB_OFS   0u                       // float2 x 16384  (32768 floats)
#define STAB_OFS   32768u                   // 8192 floats
#define XIODD_OFS  40960u                   // 256*4096 floats
#define GR_OFS     1089536u                 // 256*8192 floats
#define GI_OFS     3186688u                 // 256*8192 floats
// total = 5283840 floats = 20.2 MiB

#define PI_D 3.14159265358979323846

typedef float v2f __attribute__((ext_vector_type(2)));
typedef float v8f __attribute__((ext_vector_type(8)));

// V_WMMA_F32_16X16X4_F32: D(16x16,f32) = A(16x4,f32) * B(4x16,f32) + C
#define WMMA_F32X4(a, b, c) \
    __builtin_amdgcn_wmma_f32_16x16x4_f32(false, (a), false, (b), (short)0, (c), false, false)

// ---------------------------------------------------------------------------
// Kernel 0: chirp-conv table w[d]=sqrt(c/pi)*exp(i*c*d^2), d in [-8190,8190],
// and sinc table s[d]=(-1)^d/(pi*(d+0.5)), d in [-4095,4095]. f64 phase reduce.
// ---------------------------------------------------------------------------
__global__ __launch_bounds__(256) void frft_tables(const float* __restrict__ order_p,
                                                   float* __restrict__ ws) {
    int i = blockIdx.x * 256 + threadIdx.x;
    double a     = (double)order_p[0];
    double alpha = a * PI_D * 0.5;
    double sina  = sin(alpha);
    double c     = PI_D / (4.0 * (double)NN * sina);
    double scale = sqrt(fabs(c) / PI_D);

    if (i < WLEN) {
        float re = 0.0f, im = 0.0f;
        if (i <= 16380) {
            long long d = (long long)i - 8190;
            double ph = fmod(c * (double)(d * d), 2.0 * PI_D);
            re = (float)(scale * cos(ph));
            im = (float)(scale * sin(ph));
        }
        ws[WTAB_OFS + 2u * i]     = re;
        ws[WTAB_OFS + 2u * i + 1] = im;
    } else if (i < WLEN + SLEN) {
        int k = i - WLEN;
        float v = 0.0f;
        if (k <= 8190) {
            int d = k - 4095;
            float sgn = (d & 1) ? -1.0f : 1.0f;
            v = sgn / (float)(PI_D * ((double)d + 0.5));
        }
        ws[STAB_OFS + (unsigned)k] = v;
    }
}

// ---------------------------------------------------------------------------
// Kernel 1a: odd sinc-interpolation samples as a real Toeplitz GEMM:
//   xi_odd[r][k] = sum_l x[r][l] * stab[(k-l)+4095]
// ---------------------------------------------------------------------------
__global__ __launch_bounds__(256) void frft_sinc_gemm(const float* __restrict__ x,
                                                      float* __restrict__ ws) {
    const float* __restrict__ stab = ws + STAB_OFS;
    float* __restrict__ xiodd      = ws + XIODD_OFS;

    const int lane  = threadIdx.x & 31;
    const int wv    = threadIdx.x >> 5;
    const int rtile = blockIdx.x * 16;
    const int k0    = (blockIdx.y * 8 + wv) * 16;

    const int mrow = lane & 15;          // A row / B column (same lane slot)
    const int kk   = (lane >> 4) << 1;   // K sub-offset {0,2}
    const int kcol = k0 + mrow;

    const float* __restrict__ xr = x + (size_t)(rtile + mrow) * NN;

    v8f acc = {};
    for (int l0 = 0; l0 < NN; l0 += 4) {
        v2f aF, bF;
        const int l = l0 + kk;
        aF.x = xr[l];
        aF.y = xr[l + 1];
        const int d0 = kcol - l + 4095;  // in [0,8190] always
        bF.x = stab[d0];
        bF.y = stab[d0 - 1];
        acc = WMMA_F32X4(aF, bF, acc);
    }

    const int col   = k0 + (lane & 15);
    const int rbase = rtile + ((lane >> 4) << 3);
#pragma unroll
    for (int v = 0; v < 8; ++v)
        xiodd[(size_t)(rbase + v) * NN + col] = acc[v];
}

// ---------------------------------------------------------------------------
// Kernel 1b: chirp premultiply into split planes:
//   g[j] = exp(-i*(pi/N)*(tan(alpha/2)/4)*n^2) * xi[j],  n = j-(N-1)
//   xi[2l] = x[l], xi[2k+1] = xi_odd[k], g[8191] = 0 pad
// ---------------------------------------------------------------------------
__global__ __launch_bounds__(256) void frft_chirp_mod(const float* __restrict__ x,
                                                      const float* __restrict__ order_p,
                                                      float* __restrict__ ws) {
    const int t = blockIdx.x * 256 + threadIdx.x;   // 256*8192 threads
    const int r = t >> 13;
    const int j = t & (JP - 1);

    double a     = (double)order_p[0];
    double alpha = a * PI_D * 0.5;
    double tana2 = tan(alpha * 0.5);
    double cc    = -(PI_D / (double)NN) * (tana2 * 0.25);

    float gr = 0.0f, gi = 0.0f;
    if (j < 2 * NN - 1) {
        const float xi = (j & 1) ? ws[XIODD_OFS + (size_t)r * NN + (j >> 1)]
                                 : x[(size_t)r * NN + (j >> 1)];
        const int n = j - (NN - 1);
        const double ph = fmod(cc * (double)((long long)n * n), 2.0 * PI_D);
        float s, cph;
        __sincosf((float)ph, &s, &cph);
        gr = cph * xi;
        gi = s * xi;
    }
    ws[GR_OFS + (size_t)r * JP + j] = gr;
    ws[GI_OFS + (size_t)r * JP + j] = gi;
}

// ---------------------------------------------------------------------------
// Kernel 2: complex Toeplitz chirp-convolution GEMM + fused epilogue:
//   out[r][k] = E * chrp[2k-(N-1)] * sum_j w[2k-j] * g[r][j]
// Per K-chunk the block's d-window (766 complex entries, 6 KB) is staged into
// LDS with GLOBAL_LOAD_ASYNC_TO_LDS_B64 (ASYNCcnt), then B fragments come from
// ds_load_b64. Each wave computes two 16x16 tiles sharing A fragments.
// ---------------------------------------------------------------------------
__global__ __launch_bounds__(256) void frft_chirp_gemm(const float* __restrict__ order_p,
                                                       const float* __restrict__ ws,
                                                       float* __restrict__ out) {
    __shared__ float2 wsm[SPAN];

    const float2* __restrict__ wtab = (const float2*)(ws + WTAB_OFS);
    const float* __restrict__ Gr    = ws + GR_OFS;
    const float* __restrict__ Gi    = ws + GI_OFS;

    const int lane  = threadIdx.x & 31;
    const int wv    = threadIdx.x >> 5;
    const int rtile = blockIdx.x * 16;
    const int kb    = blockIdx.y * 256;        // block column base (8 waves x 32)
    const int k0    = kb + wv * 32;            // wave's two column tiles

    const int mrow = lane & 15;
    const int kk   = (lane >> 4) << 1;
    const int kc0  = k0 + mrow;                // B column, tile 0 (tile 1 = +16)

    const float* __restrict__ grp = Gr + (size_t)(rtile + mrow) * JP;
    const float* __restrict__ gip = Gi + (size_t)(rtile + mrow) * JP;

    v8f acc_rr0 = {}, acc_ii0 = {}, acc_ri0 = {}, acc_ir0 = {};
    v8f acc_rr1 = {}, acc_ii1 = {}, acc_ri1 = {}, acc_ir1 = {};

    for (int j0 = 0; j0 < JP; j0 += TJ) {
        const int dbase = 2 * kb - (j0 + TJ - 1) + 8190;  // window start (may be <0)

        // ---- async stage: wtab[dbase .. dbase+765] -> LDS -----------------
        for (int t = threadIdx.x; t < 766; t += 256) {
            int gidx = dbase + t;
            gidx = gidx < 0 ? 0 : (gidx > WLEN - 1 ? WLEN - 1 : gidx);
            unsigned lds_addr = (unsigned)(uintptr_t)(&wsm[t]);
            unsigned long long gaddr = (unsigned long long)(uintptr_t)(wtab + gidx);
            asm volatile("global_load_async_to_lds_b64 %0, %1, off"
                         :: "v"(lds_addr), "v"(gaddr) : "memory");
        }
        asm volatile("s_wait_asynccnt 0x0" ::: "memory");
        __builtin_amdgcn_s_barrier();

        // ---- compute on the chunk ----------------------------------------
        for (int jj = 0; jj < TJ; jj += 4) {
            const int j = j0 + jj + kk;
            v2f ar, ai;
            ar.x = grp[j];
            ar.y = grp[j + 1];
            ai.x = gip[j];
            ai.y = gip[j + 1];

            // tile 0: LDS index for K-row j is (2*kc0 - j + 8190) - dbase
            const int t0 = 2 * kc0 - j + 8190 - dbase;
            const float2 w0 = wsm[t0];
            const float2 w1 = wsm[t0 - 1];
            v2f br0, bi0;
            br0.x = w0.x; br0.y = w1.x;
            bi0.x = w0.y; bi0.y = w1.y;
            acc_rr0 = WMMA_F32X4(ar, br0, acc_rr0);
            acc_ii0 = WMMA_F32X4(ai, bi0, acc_ii0);
            acc_ri0 = WMMA_F32X4(ar, bi0, acc_ri0);
            acc_ir0 = WMMA_F32X4(ai, br0, acc_ir0);

            // tile 1: columns +16  ->  d index +32
            const float2 w2 = wsm[t0 + 32];
            const float2 w3 = wsm[t0 + 31];
            v2f br1, bi1;
            br1.x = w2.x; br1.y = w3.x;
            bi1.x = w2.y; bi1.y = w3.y;
            acc_rr1 = WMMA_F32X4(ar, br1, acc_rr1);
            acc_ii1 = WMMA_F32X4(ai, bi1, acc_ii1);
            acc_ri1 = WMMA_F32X4(ar, bi1, acc_ri1);
            acc_ir1 = WMMA_F32X4(ai, br1, acc_ir1);
        }
        __builtin_amdgcn_s_barrier();   // protect LDS window before next stage
    }

    // ---- epilogue: E * chrp[m], m = 2k-(N-1) ------------------------------
    double a     = (double)order_p[0];
    double alpha = a * PI_D * 0.5;
    double tana2 = tan(alpha * 0.5);
    double cc    = -(PI_D / (double)NN) * (tana2 * 0.25);

    const int rbase = rtile + ((lane >> 4) << 3);
    float2* __restrict__ outc = (float2*)out;

#pragma unroll
    for (int tcol = 0; tcol < 2; ++tcol) {
        const int col = k0 + tcol * 16 + (lane & 15);
        const int m   = 2 * col - (NN - 1);
        const double ph = fmod(cc * (double)((long long)m * m) - (1.0 - a) * PI_D * 0.25,
                               2.0 * PI_D);
        float es, ec;
        __sincosf((float)ph, &es, &ec);

        const v8f& rr = tcol ? acc_rr1 : acc_rr0;
        const v8f& ii = tcol ? acc_ii1 : acc_ii0;
        const v8f& ri = tcol ? acc_ri1 : acc_ri0;
        const v8f& ir = tcol ? acc_ir1 : acc_ir0;
#pragma unroll
        for (int v = 0; v < 8; ++v) {
            const float re = rr[v] - ii[v];
            const float im = ri[v] + ir[v];
            float2 o;
            o.x = ec * re - es * im;
            o.y = ec * im + es * re;
            outc[(size_t)(rbase + v) * NN + col] = o;
        }
    }
}

// ---------------------------------------------------------------------------
extern "C" void kernel_launch(void* const* d_in, const int* in_sizes, int n_in,
                              void* d_out, int out_size, void* d_ws, size_t ws_size,
                              hipStream_t stream) {
    const float* x     = (const float*)d_in[0];
    const float* order = (const float*)d_in[1];
    float*       out   = (float*)d_out;
    float*       ws    = (float*)d_ws;

    // 0) tables (w chirp table + sinc table), f64 phase reduction on device
    frft_tables<<<(WLEN + SLEN + 255) / 256, 256, 0, stream>>>(order, ws);

    // 1a) odd sinc samples: real Toeplitz GEMM on WMMA f32
    frft_sinc_gemm<<<dim3(16, 32), 256, 0, stream>>>(x, ws);

    // 1b) chirp premultiply into split Gr/Gi planes (padded)
    frft_chirp_mod<<<(RR * JP) / 256, 256, 0, stream>>>(x, order, ws);

    // 2) complex Toeplitz chirp GEMM, LDS-staged B window, fused post-chirp
    frft_chirp_gemm<<<dim3(16, 16), 256, 0, stream>>>(order, ws, out);
}